// CRFHead_27891517620990
// MI455X (gfx1250) — compile-verified
//
#include <hip/hip_runtime.h>
#include <math.h>

// ---- problem constants (match reference) ----
#define B_ 2
#define H_ 96
#define W_ 96
#define N_ (H_ * W_)          // 9216 pixels per image
#define NT_ (N_ / 16)         // 576 16-wide tiles
#define SRGB_ 5.0f
#define SCOMP_ 5.0f
#define SXY_ 60.0f
#define CRF_SCALE_ 0.7f
#define REFINE_ITERS_ 10
#define LOG2E_ 1.4426950408889634f
#define SQRT_LOG2E_ 1.2011224087864498f   // sqrt(log2(e)) folded into features

typedef __attribute__((ext_vector_type(2))) float v2f;
typedef __attribute__((ext_vector_type(8))) float v8f;

// Raw v_exp_f32 (exp2). Arguments are <= ~1e-3 (mathematically <= 0, fp noise
// only); deep-negative inputs flush to zero in hardware, which is exactly the
// desired kernel behavior — no libm denorm fixup, no canonicalize, no clamp.
__device__ __forceinline__ float fast_exp2(float x) {
    return __builtin_amdgcn_exp2f(x);
}

// -------------------------------------------------------------------------
// Augmented-feature Gram tile via FP32 WMMA.
// featA row i = [f_i*sqrt(log2e) (5), -sqh_i, 1, 0]
// featB row j = [f_j*sqrt(log2e) (5), 1, -sqh_j, 0]
// Two V_WMMA_F32_16X16X4_F32 (K=0..3, K=4..7) produce directly
//   t_ij = log2e*(f_i.f_j - 0.5|f_i|^2 - 0.5|f_j|^2) = -0.5*log2e*d2_ij
// so the kernel value is exp2(t) with zero VALU bias work.
// f32 16x4 A and 4x16 B share the same lane layout: lane holds row (lane&15),
// VGPR v at half h holds K = 2*h + v (+4 for the second WMMA).
// -------------------------------------------------------------------------
__device__ __forceinline__ v8f gram_tile(const float* __restrict__ featB,
                                         v2f a0, v2f a1, int jb,
                                         int col, int half) {
    const float* fb = featB + (size_t)(jb + col) * 8 + 2 * half;
    v2f b0; b0.x = fb[0]; b0.y = fb[1];
    v2f b1; b1.x = fb[4]; b1.y = fb[5];
    v8f c = {};
    c = __builtin_amdgcn_wmma_f32_16x16x4_f32(false, a0, false, b0,
                                              (short)0, c, false, false);
    c = __builtin_amdgcn_wmma_f32_16x16x4_f32(false, a1, false, b1,
                                              (short)0, c, false, false);
    return c;
}

__device__ __forceinline__ float half16_sum(float v) {
    // sum across the 16 lanes of each wave half (xor of bits 0..3 only)
    v += __shfl_xor(v, 1, 32);
    v += __shfl_xor(v, 2, 32);
    v += __shfl_xor(v, 4, 32);
    v += __shfl_xor(v, 8, 32);
    return v;
}

// -------------------------------------------------------------------------
// Kernel 0: zero the per-image mask-max accumulators (d_ws is poisoned).
// -------------------------------------------------------------------------
__global__ void crf_init_kernel(int* __restrict__ maxint) {
    if (threadIdx.x < B_) maxint[threadIdx.x] = 0;
}

// -------------------------------------------------------------------------
// Kernel 1: preprocessing. Quantize image/mask, build augmented features
// (bias folded into dims 5/6), per-image mask max (int atomic).
// -------------------------------------------------------------------------
__global__ void crf_prep_kernel(const float* __restrict__ imgs,
                                const float* __restrict__ masks,
                                float* __restrict__ featA,
                                float* __restrict__ featB,
                                float* __restrict__ maskq,
                                int* __restrict__ maxint) {
    int idx = blockIdx.x * blockDim.x + threadIdx.x;
    if (idx >= B_ * N_) return;
    int b = idx / N_;
    int p = idx - b * N_;
    int y = p / W_;
    int x = p - y * W_;

    const float mean[3] = {0.485f, 0.456f, 0.406f};
    const float stdv[3] = {0.229f, 0.224f, 0.225f};

    float f[5];
    f[0] = (float)x / SXY_;
    f[1] = (float)y / SXY_;
#pragma unroll
    for (int c = 0; c < 3; ++c) {
        float v = imgs[((size_t)(b * 3 + c) * H_ + y) * W_ + x];
        v = (v * stdv[c] + mean[c]) * 255.0f;
        v = floorf(fminf(fmaxf(v, 0.0f), 255.0f));   // uint8 quantization
        f[2 + c] = v / SRGB_;
    }

    float s2 = 0.0f;
#pragma unroll
    for (int k = 0; k < 5; ++k) {
        float fs = f[k] * SQRT_LOG2E_;               // fold log2e into Gram
        featA[(size_t)idx * 8 + k] = fs;
        featB[(size_t)idx * 8 + k] = fs;
        s2 += fs * fs;
    }
    float nsqh = -0.5f * s2;                         // = -0.5*log2e*|f|^2
    featA[(size_t)idx * 8 + 5] = nsqh;
    featA[(size_t)idx * 8 + 6] = 1.0f;
    featA[(size_t)idx * 8 + 7] = 0.0f;
    featB[(size_t)idx * 8 + 5] = 1.0f;
    featB[(size_t)idx * 8 + 6] = nsqh;
    featB[(size_t)idx * 8 + 7] = 0.0f;

    float mq = masks[idx] * (255.0f / CRF_SCALE_);
    mq = floorf(fminf(fmaxf(mq, 0.0f), 255.0f));
    maskq[idx] = mq;
    atomicMax(&maxint[b], (int)mq);                  // mq is a small non-neg int
}

// -------------------------------------------------------------------------
// Kernel 2: degree pass. One wave per 16-row i-block; sweep all j-tiles,
// K-tile = exp2(t) straight out of the WMMA, accumulate row sums,
// store rsq_i = 1/sqrt(deg_i).
// -------------------------------------------------------------------------
__global__ void __launch_bounds__(256)
crf_deg_kernel(const float* __restrict__ featA_all,
               const float* __restrict__ featB_all,
               float* __restrict__ rsq_all) {
    int wave = (blockIdx.x * blockDim.x + threadIdx.x) >> 5;
    int lane = threadIdx.x & 31;
    int b = wave / NT_;
    if (b >= B_) return;                              // wave-uniform
    int ib = (wave - b * NT_) * 16;

    const float* featA = featA_all + (size_t)b * N_ * 8;
    const float* featB = featB_all + (size_t)b * N_ * 8;

    int col = lane & 15;
    int half = lane >> 4;

    // Loop-invariant A operand (rows ib..ib+15), bias included.
    const float* fa = featA + (size_t)(ib + col) * 8 + 2 * half;
    v2f a0; a0.x = fa[0]; a0.y = fa[1];
    v2f a1; a1.x = fa[4]; a1.y = fa[5];

    float acc[8];
#pragma unroll
    for (int r = 0; r < 8; ++r) acc[r] = 0.0f;

#pragma unroll 2
    for (int jt = 0; jt < NT_; ++jt) {
        int jb = jt * 16;
        int jn = (jt + 1 < NT_) ? jb + 16 : 0;
        __builtin_prefetch(featB + (size_t)(jn + col) * 8, 0, 3);

        v8f c = gram_tile(featB, a0, a1, jb, col, half);
#pragma unroll
        for (int r = 0; r < 8; ++r)
            acc[r] += fast_exp2(c[r]);
    }
#pragma unroll
    for (int r = 0; r < 8; ++r) acc[r] = half16_sum(acc[r]);

    if (col == 0) {
#pragma unroll
        for (int r = 0; r < 8; ++r) {
            int i = ib + r + 8 * half;
            rsq_all[(size_t)b * N_ + i] = rsqrtf(acc[r]);
        }
    }
}

// -------------------------------------------------------------------------
// Kernel 3: unary logits l = log([1-U, U]) and the *rsq-prescaled* initial
// Q0s = [1-U, U] * rsq_i (runs after the degree pass).
// -------------------------------------------------------------------------
__global__ void crf_unary_kernel(const float* __restrict__ maskq,
                                 const int* __restrict__ maxint,
                                 const float* __restrict__ rsq,
                                 float* __restrict__ logits,
                                 float* __restrict__ Q0s) {
    int idx = blockIdx.x * blockDim.x + threadIdx.x;
    if (idx >= B_ * N_) return;
    int b = idx / N_;
    float mx = (float)maxint[b];
    float U = maskq[idx] / (mx + 1e-8f);
    U = fminf(fmaxf(U, 1e-6f), 1.0f - 1e-6f);
    logits[2 * idx + 0] = logf(1.0f - U);
    logits[2 * idx + 1] = logf(U);
    float ri = rsq[idx];
    Q0s[2 * idx + 0] = (1.0f - U) * ri;
    Q0s[2 * idx + 1] = U * ri;
}

// -------------------------------------------------------------------------
// Kernel 4: one mean-field iteration, fully fused. Qin is pre-scaled by
// rsq_j, so per tile: 2 b64 feature loads + 1 b64 Q load, 2 WMMA,
// 8x v_exp_f32, 16 FMAs. Epilogue writes the next iteration's pre-scaled Q
// (p * rsq_i); on the last iteration also writes raw p1 to out.
// -------------------------------------------------------------------------
__global__ void __launch_bounds__(256)
crf_iter_kernel(const float* __restrict__ featA_all,
                const float* __restrict__ featB_all,
                const float* __restrict__ rsq_all,
                const float* __restrict__ logits_all,
                const float* __restrict__ Qin_all,   // pre-scaled by rsq
                float* __restrict__ Qout_all,        // written pre-scaled
                float* __restrict__ out) {
    int wave = (blockIdx.x * blockDim.x + threadIdx.x) >> 5;
    int lane = threadIdx.x & 31;
    int b = wave / NT_;
    if (b >= B_) return;                              // wave-uniform
    int ib = (wave - b * NT_) * 16;

    const float* featA  = featA_all  + (size_t)b * N_ * 8;
    const float* featB  = featB_all  + (size_t)b * N_ * 8;
    const float* rsq    = rsq_all    + (size_t)b * N_;
    const float* logits = logits_all + (size_t)b * N_ * 2;
    const float* Qin    = Qin_all    + (size_t)b * N_ * 2;
    float*       Qout   = Qout_all   + (size_t)b * N_ * 2;

    int col = lane & 15;
    int half = lane >> 4;

    const float* fa = featA + (size_t)(ib + col) * 8 + 2 * half;
    v2f a0; a0.x = fa[0]; a0.y = fa[1];
    v2f a1; a1.x = fa[4]; a1.y = fa[5];

    float acc0[8], acc1[8];
#pragma unroll
    for (int r = 0; r < 8; ++r) { acc0[r] = 0.0f; acc1[r] = 0.0f; }

#pragma unroll 2
    for (int jt = 0; jt < NT_; ++jt) {
        int jb = jt * 16;
        int jn = (jt + 1 < NT_) ? jb + 16 : 0;
        __builtin_prefetch(featB + (size_t)(jn + col) * 8, 0, 3);

        v8f c = gram_tile(featB, a0, a1, jb, col, half);

        int j = jb + col;
        v2f q = *(const v2f*)(Qin + 2 * j);           // one b64: rsq_j*Q_j[0,1]
#pragma unroll
        for (int r = 0; r < 8; ++r) {
            float k = fast_exp2(c[r]);
            acc0[r] += k * q.x;
            acc1[r] += k * q.y;
        }
    }
#pragma unroll
    for (int r = 0; r < 8; ++r) {
        acc0[r] = half16_sum(acc0[r]);
        acc1[r] = half16_sum(acc1[r]);
    }

    if (col == 0) {
#pragma unroll
        for (int r = 0; r < 8; ++r) {
            int i = ib + r + 8 * half;
            float ri = rsq[i];
            float s = SCOMP_ * ri;
            float l0 = logits[2 * i + 0] + s * acc0[r];
            float l1 = logits[2 * i + 1] + s * acc1[r];
            // stable 2-class softmax: p1 = 1 / (1 + exp(l0 - l1))
            float p1 = 1.0f / (1.0f + fast_exp2((l0 - l1) * LOG2E_));
            Qout[2 * i + 0] = (1.0f - p1) * ri;       // pre-scale for next iter
            Qout[2 * i + 1] = p1 * ri;
            if (out) out[(size_t)b * N_ + i] = p1;
        }
    }
}

// -------------------------------------------------------------------------
// Host launcher. Workspace use: ~1.7 MB of f32 scratch carved from d_ws.
// -------------------------------------------------------------------------
extern "C" void kernel_launch(void* const* d_in, const int* in_sizes, int n_in,
                              void* d_out, int out_size, void* d_ws, size_t ws_size,
                              hipStream_t stream) {
    const float* imgs  = (const float*)d_in[0];   // (B,3,H,W) f32
    const float* masks = (const float*)d_in[1];   // (B,H,W)   f32

    float* ws = (float*)d_ws;
    float* featA  = ws;                                 // B*N*8
    float* featB  = featA  + (size_t)B_ * N_ * 8;       // B*N*8
    float* maskq  = featB  + (size_t)B_ * N_ * 8;       // B*N
    float* rsq    = maskq  + (size_t)B_ * N_;           // B*N
    float* logits = rsq    + (size_t)B_ * N_;           // B*N*2
    float* Qa     = logits + (size_t)B_ * N_ * 2;       // B*N*2 (rsq-scaled)
    float* Qb     = Qa     + (size_t)B_ * N_ * 2;       // B*N*2 (rsq-scaled)
    int*   maxint = (int*)(Qb + (size_t)B_ * N_ * 2);   // B ints

    const int nPix = B_ * N_;
    const int pixBlocks = (nPix + 255) / 256;
    const int nWaves = B_ * NT_;                        // 1152 waves
    const int tileBlocks = (nWaves * 32) / 256;         // 144 blocks, 8 waves each

    crf_init_kernel<<<1, 32, 0, stream>>>(maxint);
    crf_prep_kernel<<<pixBlocks, 256, 0, stream>>>(imgs, masks, featA, featB,
                                                   maskq, maxint);
    crf_deg_kernel<<<tileBlocks, 256, 0, stream>>>(featA, featB, rsq);
    crf_unary_kernel<<<pixBlocks, 256, 0, stream>>>(maskq, maxint, rsq,
                                                    logits, Qa);

    float* qin = Qa;
    float* qout = Qb;
    for (int it = 0; it < REFINE_ITERS_; ++it) {
        float* outp = (it == REFINE_ITERS_ - 1) ? (float*)d_out : nullptr;
        crf_iter_kernel<<<tileBlocks, 256, 0, stream>>>(featA, featB, rsq,
                                                        logits, qin, qout, outp);
        float* t = qin; qin = qout; qout = t;
    }
}